// Filter_21534966022815
// MI455X (gfx1250) — compile-verified
//
#include <hip/hip_runtime.h>
#include <hip/hip_bf16.h>

#define NV 1024
#define NNE (NV * NV)          // 1M elements per matrix
#define NE 32768
#define NF 4
#define MORD 33                // CHEB_ORDER + 1 coefficients
#define BT 64                  // block tile (M and N)
#define KB 64                  // K slab staged through LDS
#define KBP 68                 // padded LDS row stride (floats): %4==0 for b128, bank-safe
#define NSLAB (NV / KB)
#define PI_F 3.14159265358979323846f

typedef float v2f __attribute__((ext_vector_type(2)));
typedef float v8f __attribute__((ext_vector_type(8)));
typedef unsigned int u32x4 __attribute__((ext_vector_type(4)));
typedef int i32x4 __attribute__((ext_vector_type(4)));
typedef int i32x8 __attribute__((ext_vector_type(8)));

#if defined(__has_builtin)
#if __has_builtin(__builtin_amdgcn_tensor_load_to_lds) && \
    __has_builtin(__builtin_amdgcn_s_wait_tensorcnt)
#define USE_TDM 1
#endif
#endif

// ---------------- setup kernels ----------------

__global__ void fill_zero_kernel(float* __restrict__ p, int n) {
    int i = blockIdx.x * blockDim.x + threadIdx.x;
    if (i < n) p[i] = 0.0f;
}

__global__ void scatter_edges_kernel(const float* __restrict__ w,
                                     const int* __restrict__ ei,
                                     float* __restrict__ A, int ne) {
    int e = blockIdx.x * blockDim.x + threadIdx.x;
    if (e < ne) {
        int src = ei[e];
        int dst = ei[ne + e];
        A[(size_t)src * NV + dst] = w[e];
    }
}

__global__ void sub_diag_kernel(float* __restrict__ A) {
    int i = blockIdx.x * blockDim.x + threadIdx.x;
    if (i < NV) A[(size_t)i * (NV + 1)] -= 1.0f;
}

__global__ void init_identity_kernel(float* __restrict__ P) {
    int i = blockIdx.x * blockDim.x + threadIdx.x;
    if (i < NNE) P[i] = ((i % (NV + 1)) == 0) ? 1.0f : 0.0f;
}

__global__ void cheb_coeff_kernel(const float* __restrict__ taus,
                                  float* __restrict__ c) {
    int o = threadIdx.x;   // 0..32
    int f = threadIdx.y;   // 0..3
    if (o >= MORD || f >= NF) return;
    float tau = taus[f];
    float s = 0.0f;
    for (int n = 0; n < MORD; ++n) {
        float ph  = PI_F * ((float)n + 0.5f) / (float)MORD;
        float lam = __cosf(ph) + 1.0f;               // a1*pt + a2, a1=a2=1
        s += __cosf(ph * (float)o) * __expf(-tau * lam);
    }
    c[o * NF + f] = (2.0f / (float)MORD) * s;
}

__global__ void init_r_kernel(const float* __restrict__ T1,
                              const float* __restrict__ c,
                              float* __restrict__ R) {
    long long i = (long long)blockIdx.x * blockDim.x + threadIdx.x;
    if (i >= (long long)NF * NNE) return;
    int f   = (int)(i >> 20);          // NNE == 1<<20
    int off = (int)(i & (NNE - 1));
    float v = c[1 * NF + f] * T1[off];
    if ((off % (NV + 1)) == 0) v += 0.5f * c[0 * NF + f];
    R[i] = v;
}

// ---------------- fused Chebyshev-step GEMM (f32 WMMA + TDM staging) ----------------
// acc = F @ C   (F = T1, left operand every step)
// Tnew = 2*acc - Told ;  r[f] += coeff[k,f] * Tnew
__global__ __launch_bounds__(128, 2)
void cheb_gemm_wmma(const float* __restrict__ F,
                    const float* __restrict__ Cm,
                    const float* __restrict__ O,
                    float* __restrict__ D,
                    float* __restrict__ R,
                    const float* __restrict__ coeff,
                    int k) {
    __shared__ float As[2][BT][KBP];   // rows m, cols k; stride 68 -> bank (4m+k), b128-alignable
    __shared__ float Bs[2][KB][KBP];   // rows k, cols n

    const int tid  = threadIdx.x;
    const int lane = tid & 31;         // wave32
    const int w    = tid >> 5;         // 4 waves, 2x2 grid of 32x32 wave tiles
    const int half = lane >> 4;
    const int l16  = lane & 15;
    const int wm0  = (w >> 1) * 32;
    const int wn0  = (w & 1) * 32;
    const int bm0  = blockIdx.y * BT;
    const int bn0  = blockIdx.x * BT;

    v8f acc[2][2] = {};

    auto compute_slab = [&](const float (*pA)[KBP], const float (*pB)[KBP]) {
        #pragma unroll
        for (int k4 = 0; k4 < KB; k4 += 4) {
            v2f a[2], b[2];
            #pragma unroll
            for (int mi = 0; mi < 2; ++mi) {
                int m = wm0 + 16 * mi + l16;
                a[mi].x = pA[m][k4 + 2 * half];
                a[mi].y = pA[m][k4 + 2 * half + 1];
            }
            #pragma unroll
            for (int ni = 0; ni < 2; ++ni) {
                int n = wn0 + 16 * ni + l16;
                b[ni].x = pB[k4 + 2 * half][n];
                b[ni].y = pB[k4 + 2 * half + 1][n];
            }
            #pragma unroll
            for (int mi = 0; mi < 2; ++mi)
                #pragma unroll
                for (int ni = 0; ni < 2; ++ni)
                    acc[mi][ni] = __builtin_amdgcn_wmma_f32_16x16x4_f32(
                        false, a[mi], false, b[ni], (short)0, acc[mi][ni],
                        false, false);
        }
    };

#ifdef USE_TDM
    // ---- Tensor Data Mover double-buffered pipeline (6-arg builtin, clang-23 lane) ----
    // D# group1: data_size=4B (2<<16) | pad_enable (1<<20) | pad_interval=64dw (5<<22)
    //            | pad_amount=4dw (3<<25)  => LDS row stride 68 floats
    auto tdm_issue = [&](int buf, int kk) {
        unsigned ldsA = (unsigned)(size_t)(void*)&As[buf][0][0];  // flat LDS addr[31:0] == LDS offset
        unsigned ldsB = (unsigned)(size_t)(void*)&Bs[buf][0][0];
        unsigned long long gA = (unsigned long long)(size_t)(F  + (size_t)bm0 * NV + kk);
        unsigned long long gB = (unsigned long long)(size_t)(Cm + (size_t)kk  * NV + bn0);
        u32x4 g0a = { 1u, ldsA, (unsigned)gA, (unsigned)(gA >> 32) | 0x80000000u };
        u32x4 g0b = { 1u, ldsB, (unsigned)gB, (unsigned)(gB >> 32) | 0x80000000u };
        i32x8 g1  = { (int)0x07520000,   // mask=0 | data_size=2 | pad_en | interval=5 | amount=3
                      0x04000000,        // tensor_dim0 = 1024 (bits 79:48)
                      0x04000000,        // tensor_dim1 = 1024 (bits 111:80)
                      0x00400000,        // tile_dim0   = 64   (bits 127:112)
                      64,                // tile_dim1   = 64, tile_dim2 = 0
                      1024,              // tensor_dim0_stride = 1024 (bits 207:160)
                      0, 0 };
        i32x4 gz4 = { 0, 0, 0, 0 };      // groups 2/3 unused (2D tensor)
        i32x8 gz8 = { 0, 0, 0, 0, 0, 0, 0, 0 };
        __builtin_amdgcn_tensor_load_to_lds(g0a, g1, gz4, gz4, gz8, 0);
        __builtin_amdgcn_tensor_load_to_lds(g0b, g1, gz4, gz4, gz8, 0);
    };

    if (w == 0) tdm_issue(0, 0);
    int buf = 0;
    for (int s = 0; s < NSLAB; ++s) {
        if (w == 0) __builtin_amdgcn_s_wait_tensorcnt(0);  // slab s landed in LDS
        __syncthreads();                                   // visible to all waves
        if (w == 0 && s + 1 < NSLAB) tdm_issue(buf ^ 1, (s + 1) * KB);  // prefetch next slab
        compute_slab(As[buf], Bs[buf]);
        buf ^= 1;
    }
#else
    // ---- fallback: synchronous VGPR staging ----
    for (int kk = 0; kk < NV; kk += KB) {
        __syncthreads();
        #pragma unroll
        for (int i = 0; i < 8; ++i) {
            int idx = tid + i * 128;           // 0..1023 float4 slots
            int row = idx >> 4;
            int c4  = (idx & 15) << 2;
            const float4 va = *(const float4*)(F + (size_t)(bm0 + row) * NV + kk + c4);
            *(float4*)&As[0][row][c4] = va;
            const float4 vb = *(const float4*)(Cm + (size_t)(kk + row) * NV + bn0 + c4);
            *(float4*)&Bs[0][row][c4] = vb;
        }
        __syncthreads();
        compute_slab(As[0], Bs[0]);
    }
#endif

    // epilogue: Tnew = 2*acc - Told, r[f] += c[k,f]*Tnew
    const float c0 = coeff[k * NF + 0];
    const float c1 = coeff[k * NF + 1];
    const float c2 = coeff[k * NF + 2];
    const float c3 = coeff[k * NF + 3];
    #pragma unroll
    for (int mi = 0; mi < 2; ++mi) {
        #pragma unroll
        for (int ni = 0; ni < 2; ++ni) {
            int gc = bn0 + wn0 + 16 * ni + l16;
            #pragma unroll
            for (int j = 0; j < 8; ++j) {
                int gr = bm0 + wm0 + 16 * mi + 8 * half + j;   // C/D layout: row = j + 8*half
                size_t off = (size_t)gr * NV + gc;
                float d = 2.0f * acc[mi][ni][j] - O[off];
                D[off] = d;
                R[(size_t)0 * NNE + off] += c0 * d;
                R[(size_t)1 * NNE + off] += c1 * d;
                R[(size_t)2 * NNE + off] += c2 * d;
                R[(size_t)3 * NNE + off] += c3 * d;
            }
        }
    }
}

// ---------------- host orchestration ----------------

extern "C" void kernel_launch(void* const* d_in, const int* in_sizes, int n_in,
                              void* d_out, int out_size, void* d_ws, size_t ws_size,
                              hipStream_t stream) {
    const float* edge_w  = (const float*)d_in[0];   // (32768,)
    const float* taus    = (const float*)d_in[1];   // (4,)
    const int*   edge_ix = (const int*)d_in[2];     // (2, 32768) flat
    float* R = (float*)d_out;                       // (4, 1024, 1024)

    float* ws = (float*)d_ws;
    float* WF = ws;              // T1 (= factor/2), left operand every step
    float* WA = WF + NNE;        // rotating T buffers
    float* WB = WA + NNE;
    float* WC = WB + NNE;
    float* CB = WC + NNE;        // 33 x 4 Chebyshev coefficients

    fill_zero_kernel<<<(NNE + 255) / 256, 256, 0, stream>>>(WF, NNE);
    scatter_edges_kernel<<<(NE + 255) / 256, 256, 0, stream>>>(edge_w, edge_ix, WF, NE);
    sub_diag_kernel<<<(NV + 255) / 256, 256, 0, stream>>>(WF);
    init_identity_kernel<<<(NNE + 255) / 256, 256, 0, stream>>>(WA);
    cheb_coeff_kernel<<<1, dim3(MORD, NF), 0, stream>>>(taus, CB);
    init_r_kernel<<<((long long)NF * NNE + 255) / 256, 256, 0, stream>>>(WF, CB, R);

    // Chebyshev recursion: k = 2..32
    const float* pold = WA;      // T0 = I
    const float* pcur = WF;      // T1
    float* pool[3] = { WB, WC, WA };
    int bi = 0;
    dim3 grid(NV / BT, NV / BT);  // 16 x 16 blocks
    for (int k = 2; k <= 32; ++k) {
        float* pnew = pool[bi];
        bi = (bi + 1) % 3;
        cheb_gemm_wmma<<<grid, 128, 0, stream>>>(WF, pcur, pold, pnew, R, CB, k);
        pold = pcur;
        pcur = pnew;
    }
    (void)n_in; (void)in_sizes; (void)out_size; (void)ws_size;
}